// LlamaAttention_78752520339541
// MI455X (gfx1250) — compile-verified
//
#include <hip/hip_runtime.h>
#include <cmath>
#include <cstdint>

typedef float v2f __attribute__((ext_vector_type(2)));
typedef float v8f __attribute__((ext_vector_type(8)));

namespace {
constexpr int kB     = 32;
constexpr int kHid   = 4096;
constexpr int kNH    = 32;
constexpr int kNKV   = 8;
constexpr int kHD    = 128;
constexpr int kG     = kNH / kNKV;             // 4
constexpr int kBS    = 16;
constexpr int kMaxB  = 128;
constexpr int kS     = kMaxB * kBS;            // 2048
constexpr int kQKVN  = (kNH + 2 * kNKV) * kHD; // 6144
constexpr int kKSplit = 8;                     // GEMM K split
constexpr int kSSplit = 8;                     // flash-decode sequence split
constexpr int kSplitLen = kS / kSSplit;        // 256 slots per split
constexpr float kScale = 0.08838834764831845f; // 1/sqrt(128)
}

__device__ __forceinline__ v8f wmma_f32(v2f a, v2f b, v8f c) {
  // D = A(16x4 f32) * B(4x16 f32) + C(16x16 f32)
  return __builtin_amdgcn_wmma_f32_16x16x4_f32(false, a, false, b, (short)0, c,
                                               false, false);
}

// One async DMA: 32 lanes x 16B = 512B global -> LDS, tracked by ASYNCcnt.
__device__ __forceinline__ void async_load_b128(uint32_t lds_byte_off,
                                                const void* gptr) {
  asm volatile("global_load_async_to_lds_b128 %0, %1, off"
               :
               : "v"(lds_byte_off), "v"((uint64_t)(uintptr_t)gptr)
               : "memory");
}
__device__ __forceinline__ void wait_async_le16() {
  asm volatile("s_wait_asynccnt 16" ::: "memory");
}
__device__ __forceinline__ void wait_async_0() {
  asm volatile("s_wait_asynccnt 0" ::: "memory");
}

// C_part[kc] = A[32 x K-slice] * B[K-slice x N].  One wave owns a 32x16 tile
// (two 16x16 M-tiles share the B operand).  K split across gridDim.y.
__global__ __launch_bounds__(256) void gemm32_wmma(
    const float* __restrict__ A, const float* __restrict__ Bm,
    float* __restrict__ Cpart, int N, int K) {
  const int lane  = threadIdx.x & 31;
  const int wave  = threadIdx.x >> 5;
  const int ntile = blockIdx.x * 8 + wave;
  if (ntile * 16 >= N) return;                 // wave-uniform exit
  const int row  = lane & 15;
  const int koff = (lane >> 4) * 2;            // lanes 16-31 hold K=2,3
  const int kc   = blockIdx.y;
  const int kLen = K / kKSplit;
  const int k0   = kc * kLen;
  const int col0 = ntile * 16 + (lane & 15);

  const float* a0 = A + (size_t)row * K + k0 + koff;
  const float* a1 = A + (size_t)(row + 16) * K + k0 + koff;
  const float* bp = Bm + (size_t)(k0 + koff) * N + col0;

  v8f c0 = {}, c1 = {};
  for (int k = 0; k < kLen; k += 4) {
    if ((k & 127) == 0)                        // stream weights into L2 ahead
      __builtin_prefetch(bp + (size_t)(k + 128) * N, 0, 1);
    v2f av0, av1, bv;
    av0.x = a0[k];     av0.y = a0[k + 1];
    av1.x = a1[k];     av1.y = a1[k + 1];
    bv.x  = bp[(size_t)k * N];
    bv.y  = bp[(size_t)k * N + N];
    c0 = wmma_f32(av0, bv, c0);
    c1 = wmma_f32(av1, bv, c1);
  }

  float* out = Cpart + (size_t)kc * kB * N;
  const int rb = (lane >> 4) * 8;              // D: lanes 16-31 hold M=8..15
  for (int r = 0; r < 8; ++r) {
    out[(size_t)(rb + r) * N + col0]      = c0[r];
    out[(size_t)(16 + rb + r) * N + col0] = c1[r];
  }
}

__global__ void reduce_ksplit(const float* __restrict__ part,
                              float* __restrict__ out, int total, int stride) {
  int i = blockIdx.x * blockDim.x + threadIdx.x;
  if (i >= total) return;
  float s = 0.f;
  for (int kc = 0; kc < kKSplit; ++kc) s += part[(size_t)kc * stride + i];
  out[i] = s;
}

// In-place RoPE on the q (heads 0..31) and k (heads 32..39) sections of qkv.
__global__ void rope_kernel(float* __restrict__ qkv,
                            const int* __restrict__ positions) {
  const int b = blockIdx.x;
  const int h = blockIdx.y;
  const int i = threadIdx.x;                   // 0..63
  const float pos = (float)positions[b];
  const float inv = powf(500000.0f, -(float)i * (1.0f / 64.0f));
  const float f  = pos * inv;
  const float cs = cosf(f), sn = sinf(f);
  float* base = qkv + (size_t)b * kQKVN + (size_t)h * kHD;
  const float x1 = base[i], x2 = base[i + 64];
  base[i]      = x1 * cs - x2 * sn;
  base[i + 64] = x2 * cs + x1 * sn;
}

// Flash-decode partial: one WG (2 wave32) per (batch, kv-head, seq-split of
// 256 slots).  K chunks (16 slots x 128 dims = 16 rows of 512B) are staged
// into LDS with double-buffered global_load_async_to_lds_b128 and consumed by
// V_WMMA_F32_16X16X4_F32 (4 grouped-query rows replicated into the 16-row A
// tile).  Emits unnormalized (m, l, acc) partials; attn_merge combines.
__global__ __launch_bounds__(64) void attn_partial(
    const float* __restrict__ qkv, const float* __restrict__ Kc,
    const float* __restrict__ Vc, const int* __restrict__ positions,
    const int* __restrict__ block_tables, const int* __restrict__ out_loc,
    float* __restrict__ pm, float* __restrict__ pl,
    float* __restrict__ pacc) {
  __shared__ float kbuf[2][2][kBS * kHD];      // [wave][buf][slot*128+d] 32KB
  __shared__ float sc[kG][kSplitLen];          // 4KB scores/probs
  __shared__ float qs[kG][kHD];                // 2KB query tile

  const int b     = blockIdx.x;
  const int kv    = blockIdx.y;
  const int split = blockIdx.z;
  const int lane  = threadIdx.x & 31;
  const int wave  = threadIdx.x >> 5;
  const int pos   = positions[b];
  const int loc   = out_loc[b];
  const int col   = lane & 15;
  const int koff  = (lane >> 4) * 2;
  const int g     = col & 3;

  const size_t pidx = ((size_t)(b * kNKV + kv) * kSSplit + split) * kG;

  // Number of 16-slot chunks in this split containing any unmasked position.
  const int sbase = split * kSplitLen;
  const int valid = min(kSplitLen, pos - sbase + 1);
  const int cnum  = (valid <= 0) ? 0 : ((valid + kBS - 1) / kBS);

  if (cnum == 0) {                             // fully masked split: zeros
    if (threadIdx.x < kG) {
      pm[pidx + threadIdx.x] = -INFINITY;
      pl[pidx + threadIdx.x] = 0.f;
    }
    float* pa = pacc + pidx * kHD;
    for (int i = threadIdx.x; i < kG * kHD; i += 64) pa[i] = 0.f;
    return;
  }

  const float* qkv_b = qkv + (size_t)b * kQKVN;
  const float* knew  = qkv_b + kNH * kHD + kv * kHD;
  const float* vnew  = qkv_b + (kNH + kNKV) * kHD + kv * kHD;

  for (int idx = threadIdx.x; idx < kG * kHD; idx += 64)
    qs[idx >> 7][idx & 127] = qkv_b[(kv * kG + (idx >> 7)) * kHD + (idx & 127)];
  __syncthreads();

  const uint32_t kbb = (uint32_t)(uintptr_t)&kbuf[wave][0][0];

  auto issue_chunk = [&](int lc, int buf) {
    const int blk = block_tables[b * kMaxB + ((sbase + lc * kBS) >> 4)];
    const uint32_t dst = kbb + (uint32_t)buf * (kBS * kHD * 4) + lane * 16;
    for (int r = 0; r < kBS; ++r) {
      const int slot = blk * kBS + r;
      const float* src =
          (slot == loc) ? knew : (Kc + ((size_t)slot * kNKV + kv) * kHD);
      async_load_b128(dst + r * (kHD * 4), (const char*)src + lane * 16);
    }
  };

  // ---- Phase 1: scores, double-buffered async K staging ----
  if (wave < cnum) issue_chunk(wave, 0);
  int buf = 0;
  for (int lc = wave; lc < cnum; lc += 2) {
    const bool more = (lc + 2 < cnum);
    if (more) {
      issue_chunk(lc + 2, buf ^ 1);
      wait_async_le16();                       // in-order: chunk `lc` resident
    } else {
      wait_async_0();
    }
    const float* kb = &kbuf[wave][buf][0];
    v8f acc = {};
    for (int kk = 0; kk < kHD; kk += 4) {
      v2f a, bv;
      a.x  = qs[g][kk + koff];
      a.y  = qs[g][kk + koff + 1];
      bv.x = kb[col * kHD + kk + koff];        // ds_load from staged tile
      bv.y = kb[col * kHD + kk + koff + 1];
      acc = wmma_f32(a, bv, acc);
    }
    if (lane < 16) {
      const int  s  = lc * kBS + lane;
      const bool ok = (sbase + s) <= pos;
      for (int r = 0; r < kG; ++r)
        sc[r][s] = ok ? acc[r] * kScale : -INFINITY;
    }
    buf ^= 1;
  }
  // masked-out chunks of this split
  for (int lc = wave; lc < kSplitLen / kBS; lc += 2)
    if (lc >= cnum && lane < 16)
      for (int r = 0; r < kG; ++r) sc[r][lc * kBS + lane] = -INFINITY;
  __syncthreads();

  // ---- Phase 2: local softmax stats, 2 rows per wave ----
  for (int gr = wave * 2; gr < wave * 2 + 2; ++gr) {
    float m = -INFINITY;
    for (int s = lane; s < kSplitLen; s += 32) m = fmaxf(m, sc[gr][s]);
    for (int o = 16; o > 0; o >>= 1) m = fmaxf(m, __shfl_xor(m, o, 32));
    float l = 0.f;
    for (int s = lane; s < kSplitLen; s += 32) {
      const float e = __expf(sc[gr][s] - m);
      sc[gr][s] = e;
      l += e;
    }
    for (int o = 16; o > 0; o >>= 1) l += __shfl_xor(l, o, 32);
    if (lane == 0) { pm[pidx + gr] = m; pl[pidx + gr] = l; }
  }
  __syncthreads();

  // ---- Phase 3: unnormalized P.V partial via WMMA ----
  const int send = cnum * kBS;
  for (int t = wave; t < kHD / 16; t += 2) {
    const int d0 = t * 16;
    v8f acc = {};
    for (int s = 0; s < send; s += 4) {
      const int blk   = block_tables[b * kMaxB + ((sbase + s) >> 4)];
      const int sk0   = s + koff;
      const int slot0 = blk * kBS + (sk0 & 15);
      const int slot1 = blk * kBS + ((sk0 + 1) & 15);
      const float* v0 =
          (slot0 == loc) ? vnew : (Vc + ((size_t)slot0 * kNKV + kv) * kHD);
      const float* v1 =
          (slot1 == loc) ? vnew : (Vc + ((size_t)slot1 * kNKV + kv) * kHD);
      v2f a, bv;
      a.x  = sc[g][sk0];
      a.y  = sc[g][sk0 + 1];
      bv.x = v0[d0 + col];
      bv.y = v1[d0 + col];
      acc = wmma_f32(a, bv, acc);
    }
    if (lane < 16)
      for (int r = 0; r < kG; ++r)
        pacc[(pidx + r) * kHD + d0 + lane] = acc[r];
  }
}

// Deterministic merge of the kSSplit flash-decode partials.
__global__ __launch_bounds__(128) void attn_merge(
    const float* __restrict__ pm, const float* __restrict__ pl,
    const float* __restrict__ pacc, float* __restrict__ attn_out) {
  const int b = blockIdx.x, kv = blockIdx.y, d = threadIdx.x;
  for (int g = 0; g < kG; ++g) {
    float M = -INFINITY;
    for (int s = 0; s < kSSplit; ++s)
      M = fmaxf(M, pm[((size_t)(b * kNKV + kv) * kSSplit + s) * kG + g]);
    float L = 0.f, O = 0.f;
    for (int s = 0; s < kSSplit; ++s) {
      const size_t idx = ((size_t)(b * kNKV + kv) * kSSplit + s) * kG + g;
      const float e = __expf(pm[idx] - M);
      L += pl[idx] * e;
      O += pacc[idx * kHD + d] * e;
    }
    attn_out[(size_t)b * kNH * kHD + (kv * kG + g) * kHD + d] = O / L;
  }
}

extern "C" void kernel_launch(void* const* d_in, const int* in_sizes, int n_in,
                              void* d_out, int out_size, void* d_ws,
                              size_t ws_size, hipStream_t stream) {
  (void)in_sizes; (void)n_in; (void)out_size; (void)ws_size;
  const float* hidden       = (const float*)d_in[0];
  const float* w_qkv        = (const float*)d_in[1];
  const float* w_o          = (const float*)d_in[2];
  const float* k_cache      = (const float*)d_in[3];
  const float* v_cache      = (const float*)d_in[4];
  const int*   positions    = (const int*)d_in[5];
  const int*   block_tables = (const int*)d_in[6];
  const int*   out_loc      = (const int*)d_in[7];
  float*       out          = (float*)d_out;

  float* ws       = (float*)d_ws;
  float* qkv      = ws;                                      // 32*6144
  float* attn_out = qkv + (size_t)kB * kQKVN;                // 32*4096
  float* part     = attn_out + (size_t)kB * kNH * kHD;       // 8*32*6144
  float* pacc     = part + (size_t)kKSplit * kB * kQKVN;     // 32*8*8*4*128
  float* pm       = pacc + (size_t)kB * kNKV * kSSplit * kG * kHD;
  float* pl       = pm + (size_t)kB * kNKV * kSSplit * kG;

  // 1) QKV projection (K split 8 ways, deterministic reduce)
  gemm32_wmma<<<dim3(kQKVN / 128, kKSplit), 256, 0, stream>>>(
      hidden, w_qkv, part, kQKVN, kHid);
  reduce_ksplit<<<(kB * kQKVN + 255) / 256, 256, 0, stream>>>(
      part, qkv, kB * kQKVN, kB * kQKVN);

  // 2) RoPE on q and k sections (in-place in workspace)
  rope_kernel<<<dim3(kB, kNH + kNKV), 64, 0, stream>>>(qkv, positions);

  // 3) flash-decode attention: 32*8*8 = 2048 WGs of partials, then merge
  attn_partial<<<dim3(kB, kNKV, kSSplit), 64, 0, stream>>>(
      qkv, k_cache, v_cache, positions, block_tables, out_loc, pm, pl, pacc);
  attn_merge<<<dim3(kB, kNKV), 128, 0, stream>>>(pm, pl, pacc, attn_out);

  // 4) output projection
  gemm32_wmma<<<dim3(kHid / 128, kKSplit), 256, 0, stream>>>(
      attn_out, w_o, part, kHid, kNH * kHD);
  reduce_ksplit<<<(kB * kHid + 255) / 256, 256, 0, stream>>>(
      part, out, kB * kHid, kB * kHid);
}